// MinDistanceDecoder_20813411516868
// MI455X (gfx1250) — compile-verified
//
#include <hip/hip_runtime.h>
#include <stdint.h>

typedef __attribute__((ext_vector_type(2))) float v2f;
typedef __attribute__((ext_vector_type(8))) float v8f;

#define KBITS       16      // message bits (k)
#define NSYM        32      // codeword length (n) == GEMM K dim
#define NBATCH      64      // batch size
#define NUM_CW      65536   // 2^k codewords
#define NUM_BLOCKS  64
#define TILES_PER_BLOCK (NUM_CW / 16 / NUM_BLOCKS)   // 64 codeword-tiles per block

// ---------------------------------------------------------------------------
// Prep: G rows -> 32-bit parity masks; llr = -4 * noisy / sigma2 (f32, exact)
// ---------------------------------------------------------------------------
__global__ void mdd_prep_kernel(const float* __restrict__ noisy,
                                const int* __restrict__ G,
                                const float* __restrict__ sigma2,
                                uint32_t* __restrict__ masks,
                                float* __restrict__ llr) {
    int tid = threadIdx.x;
    if (tid < KBITS) {
        uint32_t m = 0u;
        #pragma unroll
        for (int n = 0; n < NSYM; ++n)
            m |= ((uint32_t)(G[tid * NSYM + n] & 1)) << n;
        masks[tid] = m;
    }
    float scale = -4.0f / sigma2[0];
    for (int i = tid; i < NBATCH * NSYM; i += blockDim.x)
        llr[i] = noisy[i] * scale;
}

// ---------------------------------------------------------------------------
// Fused correlation GEMM (WMMA f32 16x16x4, K=32 in 8 steps) + local argmax.
// Wave w handles batch tile [16w, 16w+16); block sweeps 64 codeword tiles.
// A tile (16 codewords x 4 syms): lane L row M=L%16; lanes<16 K={0,1}, >=16 K={2,3}
// B tile (4 syms x 16 batches):   lane L col N=L%16; lanes<16 K={0,1}, >=16 K={2,3}
// D (16x16 f32): VGPR j at lane L -> M = j + (L>=16 ? 8 : 0), N = L%16
// ---------------------------------------------------------------------------
__global__ void mdd_score_kernel(const uint32_t* __restrict__ masks,
                                 const float* __restrict__ llr,
                                 float* __restrict__ pScore,
                                 uint32_t* __restrict__ pIdx) {
    const int lane  = threadIdx.x & 31;
    const int wave  = threadIdx.x >> 5;        // 0..3 -> batch tile
    const int col   = lane & 15;               // N (batch within tile) / M (codeword row for A)
    const int hi    = lane >> 4;               // 0: K off 0 / M rows 0..7 ; 1: K off 2 / M rows 8..15
    const int batch = wave * 16 + col;

    // G parity masks (uniform across lanes)
    uint32_t gm[KBITS];
    #pragma unroll
    for (int j = 0; j < KBITS; ++j) gm[j] = masks[j];

    // B operands: llr^T for this wave's 16 batches, all 8 K-steps, kept in VGPRs
    v2f bop[8];
    #pragma unroll
    for (int t = 0; t < 8; ++t) {
        int k0 = 4 * t + 2 * hi;
        bop[t].x = llr[batch * NSYM + k0];
        bop[t].y = llr[batch * NSYM + k0 + 1];
    }

    float    bestS = -3.0e38f;
    uint32_t bestI = 0u;

    const uint32_t tile0 = (uint32_t)blockIdx.x * TILES_PER_BLOCK;
    for (uint32_t tt = 0; tt < TILES_PER_BLOCK; ++tt) {
        const uint32_t cwBase = (tile0 + tt) * 16u;

        // codeword parity mask for this lane's A row (codeword index cwBase+col)
        const uint32_t cw = cwBase + (uint32_t)col;
        uint32_t cmask = 0u;
        #pragma unroll
        for (int j = 0; j < KBITS; ++j)
            if ((cw >> j) & 1u) cmask ^= gm[j];

        v8f acc = {};
        #pragma unroll
        for (int t = 0; t < 8; ++t) {
            const int p0 = 4 * t + 2 * hi;
            v2f aop;
            aop.x = 1.0f - 2.0f * (float)((cmask >> p0) & 1u);        // BPSK sign, exact f32
            aop.y = 1.0f - 2.0f * (float)((cmask >> (p0 + 1)) & 1u);
            acc = __builtin_amdgcn_wmma_f32_16x16x4_f32(
                      /*neg_a=*/false, aop, /*neg_b=*/false, bop[t],
                      /*c_mod=*/(short)0, acc, /*reuse_a=*/false, /*reuse_b=*/false);
        }

        // argmax update: lane holds scores for codewords cwBase + hi*8 + j, batch `batch`
        #pragma unroll
        for (int j = 0; j < 8; ++j) {
            const uint32_t idx = cwBase + (uint32_t)(hi * 8 + j);
            const float s = acc[j];
            if (s > bestS || (s == bestS && idx < bestI)) { bestS = s; bestI = idx; }
        }
    }

    // merge the two half-wave candidates for each batch column (wave32 shuffle)
    const float    oS = __shfl_xor(bestS, 16, 32);
    const uint32_t oI = (uint32_t)__shfl_xor((int)bestI, 16, 32);
    if (oS > bestS || (oS == bestS && oI < bestI)) { bestS = oS; bestI = oI; }

    if (lane < 16) {
        pScore[blockIdx.x * NBATCH + batch] = bestS;
        pIdx  [blockIdx.x * NBATCH + batch] = bestI;
    }
}

// ---------------------------------------------------------------------------
// Final merge across blocks + emit message bits (LSB-first) as f32
// ---------------------------------------------------------------------------
__global__ void mdd_reduce_kernel(const float* __restrict__ pScore,
                                  const uint32_t* __restrict__ pIdx,
                                  float* __restrict__ out) {
    const int b = threadIdx.x;
    if (b >= NBATCH) return;
    float    bestS = -3.0e38f;
    uint32_t bestI = 0u;
    for (int blk = 0; blk < NUM_BLOCKS; ++blk) {
        const float    s = pScore[blk * NBATCH + b];
        const uint32_t i = pIdx  [blk * NBATCH + b];
        if (s > bestS || (s == bestS && i < bestI)) { bestS = s; bestI = i; }
    }
    #pragma unroll
    for (int j = 0; j < KBITS; ++j)
        out[b * KBITS + j] = (float)((bestI >> j) & 1u);
}

// ---------------------------------------------------------------------------
extern "C" void kernel_launch(void* const* d_in, const int* in_sizes, int n_in,
                              void* d_out, int out_size, void* d_ws, size_t ws_size,
                              hipStream_t stream) {
    const float* noisy  = (const float*)d_in[0];   // [64,32] f32
    const int*   G      = (const int*)d_in[1];     // [16,32] i32 (0/1)
    const float* sigma2 = (const float*)d_in[2];   // [1] f32
    float*       out    = (float*)d_out;           // [64,16] f32

    char* ws = (char*)d_ws;
    uint32_t* masks  = (uint32_t*)ws;                                    // 16 u32
    float*    llr    = (float*)(ws + 64);                                // 64*32 f32
    float*    pScore = (float*)(ws + 64 + NBATCH * NSYM * 4);            // 64*64 f32
    uint32_t* pIdx   = (uint32_t*)(ws + 64 + NBATCH * NSYM * 4
                                      + NUM_BLOCKS * NBATCH * 4);        // 64*64 u32

    mdd_prep_kernel  <<<1, 256, 0, stream>>>(noisy, G, sigma2, masks, llr);
    mdd_score_kernel <<<NUM_BLOCKS, 128, 0, stream>>>(masks, llr, pScore, pIdx);
    mdd_reduce_kernel<<<1, 64, 0, stream>>>(pScore, pIdx, out);
}